// GCNNet_46634754900281
// MI455X (gfx1250) — compile-verified
//
#include <hip/hip_runtime.h>

typedef __attribute__((ext_vector_type(2))) float v2f;
typedef __attribute__((ext_vector_type(8))) float v8f;

#define NNODES 50000

// ---------------------------------------------------------------------------
// Dense GEMM  S[M x dout] = H[M x K] * W[K x dout]   (row-major, fp32)
// One wave32 computes a 16x64 strip (4 adjacent 16x16 tiles) with
// V_WMMA_F32_16X16X4_F32, reusing each A fetch across 4 WMMAs.
// A layout (16x4 f32): lane&15 = row M, lane>>4 selects K-pair {0,1}/{2,3}
// B layout (4x16 f32): lane&15 = col N, same K-pair striping
// D layout (16x16 f32): VGPR j holds rows j (lanes 0-15) and j+8 (lanes 16-31)
// ---------------------------------------------------------------------------
__global__ __launch_bounds__(256) void gemm_f32_wmma(
    const float* __restrict__ H, const float* __restrict__ W,
    float* __restrict__ S, int K, int dout, int nGroups, int totalWork)
{
    int wave = blockIdx.x * (blockDim.x >> 5) + (threadIdx.x >> 5);
    if (wave >= totalWork) return;                // uniform per-wave: EXEC stays all-1s

    int lane = threadIdx.x & 31;
    int row  = lane & 15;                          // M row (A) / N col (B,D)
    int kh   = lane >> 4;                          // 0 or 1 -> K-pair select

    int mTile  = wave / nGroups;
    int nGroup = wave - mTile * nGroups;
    int mBase  = mTile << 4;
    int nBase  = nGroup << 6;                      // 64 columns per wave

    const float* hrow = H + (size_t)(mBase + row) * K;        // A source row
    const float* wcol = W + nBase + row;                      // B col base (tile 0)

    v8f acc0 = {}, acc1 = {}, acc2 = {}, acc3 = {};
    for (int k0 = 0; k0 < K; k0 += 4) {
        int k = k0 + (kh << 1);
        v2f a = *(const v2f*)(hrow + k);                      // A[row][k..k+1]
        const float* w0 = wcol + (size_t)k * dout;
        const float* w1 = wcol + (size_t)(k + 1) * dout;
        v2f b0; b0.x = w0[0];  b0.y = w1[0];
        v2f b1; b1.x = w0[16]; b1.y = w1[16];
        v2f b2; b2.x = w0[32]; b2.y = w1[32];
        v2f b3; b3.x = w0[48]; b3.y = w1[48];
        acc0 = __builtin_amdgcn_wmma_f32_16x16x4_f32(false, a, false, b0, (short)0, acc0, false, false);
        acc1 = __builtin_amdgcn_wmma_f32_16x16x4_f32(false, a, false, b1, (short)0, acc1, false, false);
        acc2 = __builtin_amdgcn_wmma_f32_16x16x4_f32(false, a, false, b2, (short)0, acc2, false, false);
        acc3 = __builtin_amdgcn_wmma_f32_16x16x4_f32(false, a, false, b3, (short)0, acc3, false, false);
    }

    float* srow = S + (size_t)(mBase + (kh << 3)) * dout + nBase + row;
#pragma unroll
    for (int j = 0; j < 8; ++j) {
        srow[(size_t)j * dout + 0]  = acc0[j];
        srow[(size_t)j * dout + 16] = acc1[j];
        srow[(size_t)j * dout + 32] = acc2[j];
        srow[(size_t)j * dout + 48] = acc3[j];
    }
}

// ---------------------------------------------------------------------------
// acc[n][c] = bias[c]   (segment-sum target pre-seeded with bias)
// ---------------------------------------------------------------------------
__global__ __launch_bounds__(256) void init_bias(
    float* __restrict__ acc, const float* __restrict__ bias,
    int count, int cmask)
{
    int i = blockIdx.x * blockDim.x + threadIdx.x;
    if (i < count) acc[i] = bias[i & cmask];
}

// ---------------------------------------------------------------------------
// Hardware fp32 atomic add (no-return form -> STOREcnt; guaranteed
// GLOBAL_ATOMIC_ADD_F32, no CAS-loop fallback).
// ---------------------------------------------------------------------------
__device__ __forceinline__ void atomic_add_f32_hw(float* p, float v) {
    asm volatile("global_atomic_add_f32 %0, %1, off"
                 :
                 : "v"(p), "v"(v)
                 : "memory");
}

// ---------------------------------------------------------------------------
// Edge-parallel scatter: acc[dst] += sup[src] * ew  (fp32 atomics, L2-resident)
// One thread per (edge, 4-column chunk); consecutive threads share an edge.
// ---------------------------------------------------------------------------
__global__ __launch_bounds__(256) void spmm_scatter(
    const float* __restrict__ sup, const int* __restrict__ dstI,
    const int* __restrict__ srcI, const float* __restrict__ ew,
    float* __restrict__ acc, int E, int dout)
{
    int chunks = dout >> 2;
    long long tid = (long long)blockIdx.x * blockDim.x + threadIdx.x;
    long long total = (long long)E * chunks;
    if (tid >= total) return;

    int e = (int)(tid / chunks);
    int c = (int)(tid - (long long)e * chunks);
    int s = srcI[e];
    int d = dstI[e];
    float w = ew[e];

    float4 v = *((const float4*)(sup + (size_t)s * dout) + c);
    float* ap = acc + (size_t)d * dout + (c << 2);
    atomic_add_f32_hw(ap + 0, v.x * w);
    atomic_add_f32_hw(ap + 1, v.y * w);
    atomic_add_f32_hw(ap + 2, v.z * w);
    atomic_add_f32_hw(ap + 3, v.w * w);
}

// ---------------------------------------------------------------------------
// out[i] = max(in[i], 0)
// ---------------------------------------------------------------------------
__global__ __launch_bounds__(256) void relu_copy(
    const float* __restrict__ in, float* __restrict__ out, int count)
{
    int i = blockIdx.x * blockDim.x + threadIdx.x;
    if (i < count) out[i] = fmaxf(in[i], 0.0f);
}

static inline int ceil_div_ll(long long a, int b) { return (int)((a + b - 1) / b); }

extern "C" void kernel_launch(void* const* d_in, const int* in_sizes, int n_in,
                              void* d_out, int out_size, void* d_ws, size_t ws_size,
                              hipStream_t stream) {
    const int N = NNODES;
    const int E = in_sizes[2];                     // 1,600,000 edge weights
    const int dims[5] = {256, 128, 64, 128, 256};

    float* ws  = (float*)d_ws;
    float* sup = ws;                               // N*256 floats
    float* A0  = sup + (size_t)N * 256;            // N*128 floats (layers 1/3)
    float* A1  = A0  + (size_t)N * 128;            // N*256 floats (layers 2/4)
    float* out = (float*)d_out;

    for (int br = 0; br < 2; ++br) {
        const float* feat = (const float*)d_in[3 * br + 0];
        const int*   ei   = (const int*)  d_in[3 * br + 1];
        const float* ew   = (const float*)d_in[3 * br + 2];
        const int* dstI = ei;                      // edge_index row 0 = dst
        const int* srcI = ei + E;                  // edge_index row 1 = src

        const float* Wp[4];
        const float* Bp[4];
        for (int l = 0; l < 4; ++l) {
            Wp[l] = (const float*)d_in[6 + 8 * br + 2 * l];
            Bp[l] = (const float*)d_in[6 + 8 * br + 2 * l + 1];
        }

        // output slices: (p0, h1, p3, h4) concatenated flat
        float* pOut = out + (br == 0 ? 0LL        : 16000000LL);
        float* hOut = out + (br == 0 ? 3200000LL  : 19200000LL);

        const float* h = feat;
        for (int l = 0; l < 4; ++l) {
            int K = dims[l], dout = dims[l + 1];

            // 1) support = h @ W  (WMMA f32, 16x64 strip per wave)
            int nGroups = dout / 64;
            int totalWork = (N / 16) * nGroups;
            gemm_f32_wmma<<<ceil_div_ll(totalWork, 8), 256, 0, stream>>>(
                h, Wp[l], sup, K, dout, nGroups, totalWork);

            // 2) acc = bias (broadcast)
            float* acc = (l & 1) ? A1 : A0;
            int cnt = N * dout;
            init_bias<<<ceil_div_ll(cnt, 256), 256, 0, stream>>>(
                acc, Bp[l], cnt, dout - 1);

            // 3) acc[dst] += support[src] * ew   (hardware fp32 atomics)
            long long tot = (long long)E * (dout >> 2);
            spmm_scatter<<<ceil_div_ll(tot, 256), 256, 0, stream>>>(
                sup, dstI, srcI, ew, acc, E, dout);

            // 4) relu -> next activation (layers 2/4 go straight to d_out)
            float* rdest = (l == 1) ? pOut : (l == 3) ? hOut : acc;
            relu_copy<<<ceil_div_ll(cnt, 256), 256, 0, stream>>>(acc, rdest, cnt);

            h = rdest;
        }
    }
}